// ScaledDotProductAttention_73143293051614
// MI455X (gfx1250) — compile-verified
//
#include <hip/hip_runtime.h>
#include <math.h>

#define B_DIM 2
#define C_DIM 128
#define N_DIM 784
#define TEMP_INV (1.0f/11.313708498984761f)

typedef __attribute__((ext_vector_type(2))) float v2f;
typedef __attribute__((ext_vector_type(8))) float v8f;

// ---------------------------------------------------------------------------
// Kernel A: raw L1-distance scores.
// S[b,kk,qq] = (1/C) * sum_c | q[b,c,qq]/T - k[b,c,kk] |
// One 16x16 (kk x qq) tile per 256-thread block; q/k channel slabs in LDS.
// ---------------------------------------------------------------------------
__global__ __launch_bounds__(256) void l1_scores_kernel(const float* __restrict__ q,
                                                        const float* __restrict__ k,
                                                        float* __restrict__ S) {
  __shared__ float qs[C_DIM * 16];
  __shared__ float ks[C_DIM * 16];
  const int b   = blockIdx.z;
  const int qq0 = blockIdx.x * 16;
  const int kk0 = blockIdx.y * 16;
  const int t   = threadIdx.x;

  for (int i = t; i < C_DIM * 16; i += 256) {
    const int c = i >> 4, col = i & 15;
    qs[i] = q[((size_t)b * C_DIM + c) * N_DIM + qq0 + col] * TEMP_INV;
    ks[i] = k[((size_t)b * C_DIM + c) * N_DIM + kk0 + col];
  }
  __syncthreads();

  const int qq = t & 15;
  const int kk = t >> 4;
  float sum = 0.f;
  #pragma unroll 8
  for (int c = 0; c < C_DIM; ++c) {
    sum += fabsf(qs[c * 16 + qq] - ks[c * 16 + kk]);
  }
  S[((size_t)b * N_DIM + kk0 + kk) * N_DIM + qq0 + qq] = sum * (1.0f / C_DIM);
}

// ---------------------------------------------------------------------------
// Kernel B: in-place row softmax over last dim (Nq). One block per (b,kk) row.
// wave32 shfl-xor reductions + LDS cross-wave combine (8 waves).
// ---------------------------------------------------------------------------
__device__ __forceinline__ float wave_max(float v) {
  #pragma unroll
  for (int off = 16; off > 0; off >>= 1) v = fmaxf(v, __shfl_xor(v, off, 32));
  return v;
}
__device__ __forceinline__ float wave_sum(float v) {
  #pragma unroll
  for (int off = 16; off > 0; off >>= 1) v += __shfl_xor(v, off, 32);
  return v;
}

__global__ __launch_bounds__(256) void softmax_kernel(float* __restrict__ attn) {
  __shared__ float red[8];
  float* p = attn + (size_t)blockIdx.x * N_DIM;
  const int t = threadIdx.x;
  const int wid = t >> 5, lid = t & 31;

  // --- max ---
  float lm = -3.402823466e38f;
  for (int j = t; j < N_DIM; j += 256) lm = fmaxf(lm, p[j]);
  lm = wave_max(lm);
  if (lid == 0) red[wid] = lm;
  __syncthreads();
  float m = wave_max((lid < 8) ? red[lid] : -3.402823466e38f);
  __syncthreads();

  // --- sum of exp ---
  float ls = 0.f;
  for (int j = t; j < N_DIM; j += 256) ls += __expf(p[j] - m);
  ls = wave_sum(ls);
  if (lid == 0) red[wid] = ls;
  __syncthreads();
  float s = wave_sum((lid < 8) ? red[lid] : 0.f);
  const float inv = 1.0f / s;

  for (int j = t; j < N_DIM; j += 256) p[j] = __expf(p[j] - m) * inv;
}

// ---------------------------------------------------------------------------
// Kernel C: out[b,c,kk] = sum_q v[b,c,q] * attn[b,kk,q]
// = GEMM: D(16x16 c x kk tile) += A(v tile, 16x4) x B(attn^T tile, 4x16),
// fp32 WMMA, K=784 in 196 steps of 4. One wave per tile; EXEC all-ones.
//
// ISA lane layouts (wave32):
//   A 16x4 f32 : lanes 0-15 -> M=lane, VGPR{0,1}=K{0,1}; lanes 16-31 -> K{2,3}
//   B 4x16 f32 : lanes 0-15 -> N=lane, VGPR{0,1}=K{0,1}; lanes 16-31 -> K{2,3}
//   C/D 16x16  : VGPR j -> M=j (lanes 0-15) / M=j+8 (lanes 16-31), N=lane&15
// ---------------------------------------------------------------------------
__global__ __launch_bounds__(32) void wmma_out_kernel(const float* __restrict__ v,
                                                      const float* __restrict__ attn,
                                                      float* __restrict__ out) {
  const int kk0   = blockIdx.x * 16;
  const int c0    = blockIdx.y * 16;
  const int b     = blockIdx.z;
  const int lane  = threadIdx.x;
  const int m     = lane & 15;
  const int khalf = lane >> 4;

  const float* vrow = v    + ((size_t)b * C_DIM + c0  + m) * N_DIM + 2 * khalf;
  const float* arow = attn + ((size_t)b * N_DIM + kk0 + m) * N_DIM + 2 * khalf;

  v8f acc = {};
  #pragma unroll 4
  for (int q0 = 0; q0 < N_DIM; q0 += 4) {
    v2f av = *(const v2f*)(vrow + q0);   // A: v[b, c0+m, q0 + 2*khalf + {0,1}]
    v2f bv = *(const v2f*)(arow + q0);   // B: attn[b, kk0+m, q0 + 2*khalf + {0,1}]
    // 8 args: (neg_a, A, neg_b, B, c_mod, C, reuse_a, reuse_b)
    acc = __builtin_amdgcn_wmma_f32_16x16x4_f32(false, av, false, bv,
                                                (short)0, acc, false, false);
  }

  float* orow = out + ((size_t)b * C_DIM + c0 + 8 * khalf) * N_DIM + kk0 + m;
  #pragma unroll
  for (int i = 0; i < 8; ++i) {
    orow[(size_t)i * N_DIM] = acc[i];   // row c0 + i + 8*khalf, col kk0 + m
  }
}

// ---------------------------------------------------------------------------
extern "C" void kernel_launch(void* const* d_in, const int* in_sizes, int n_in,
                              void* d_out, int out_size, void* d_ws, size_t ws_size,
                              hipStream_t stream) {
  const float* q = (const float*)d_in[0];
  const float* k = (const float*)d_in[1];
  const float* v = (const float*)d_in[2];
  float* out  = (float*)d_out;                         // [B, C, N]
  float* attn = out + (size_t)B_DIM * C_DIM * N_DIM;   // [B, N, N] (tuple elem 2)

  l1_scores_kernel<<<dim3(N_DIM / 16, N_DIM / 16, B_DIM), 256, 0, stream>>>(q, k, attn);
  softmax_kernel<<<dim3(B_DIM * N_DIM), 256, 0, stream>>>(attn);
  wmma_out_kernel<<<dim3(N_DIM / 16, C_DIM / 16, B_DIM), 32, 0, stream>>>(v, attn, out);
}